// TemporalAttention_68891275427958
// MI455X (gfx1250) — compile-verified
//
#include <hip/hip_runtime.h>

typedef float v2f __attribute__((ext_vector_type(2)));
typedef float v8f __attribute__((ext_vector_type(8)));

#define BB 2
#define T  128
#define NN 64
#define DD 128
#define HH 8
#define DK 16
#define DE 32

// ---------------------------------------------------------------------------
// Kernel 1: qkv = x @ W_qkv + b_qkv, scatter-stored as q/k/v [bn][h][t][dk]
// x row r = bn*T + t  maps to node[b][t][n][:]  (bn = b*N + n)
// Grid: (6, 1024); block 128 threads = 4 waves; wave w -> 16x16 output tile.
// x-tile is staged into LDS with CDNA5 async-to-LDS copies (ASYNCcnt path).
// ---------------------------------------------------------------------------
__global__ __launch_bounds__(128) void qkv_kernel(
    const float* __restrict__ node,
    const float* __restrict__ Wqkv, const float* __restrict__ bqkv,
    float* __restrict__ qW, float* __restrict__ kW, float* __restrict__ vW)
{
    __shared__ float xs[16 * 128];          // 8 KB x-tile (LDS offset 0)
    const int tid  = threadIdx.x;
    const int lane = tid & 31;
    const int wave = tid >> 5;
    const int r0    = blockIdx.y * 16;      // row tile (stays within one bn)
    const int bn    = r0 >> 7;
    const int tbase = r0 & 127;
    const int b = bn >> 6, n = bn & 63;

    // async copy of the 16x128 x-tile straight into LDS (no VGPR staging)
    {
        int row = tid >> 3;
        int col = (tid & 7) * 16;
        const float* src = node + ((size_t)(b * T + tbase + row) * NN + n) * DD + col;
        unsigned ldsb = (unsigned)((row * 128 + col) * 4);   // xs @ LDS offset 0
        #pragma unroll
        for (int i = 0; i < 16; i += 4) {
            asm volatile("global_load_async_to_lds_b128 %0, %1, off"
                         :: "v"(ldsb + (unsigned)(i * 4)), "v"(src + i)
                         : "memory");
        }
        asm volatile("s_wait_asynccnt 0x0" ::: "memory");
    }
    __syncthreads();

    const int jt    = blockIdx.x * 4 + wave;   // output col tile 0..23
    const int row16 = lane & 15;
    const int khalf = 2 * (lane >> 4);
    const int jcol  = jt * 16 + row16;

    v8f c = {};
    #pragma unroll
    for (int kk = 0; kk < 128; kk += 4) {
        v2f a, bm;
        a.x  = xs[row16 * 128 + kk + khalf];
        a.y  = xs[row16 * 128 + kk + khalf + 1];
        bm.x = Wqkv[(kk + khalf) * 384 + jcol];
        bm.y = Wqkv[(kk + khalf + 1) * 384 + jcol];
        c = __builtin_amdgcn_wmma_f32_16x16x4_f32(false, a, false, bm,
                                                  (short)0, c, false, false);
    }

    const float bias = bqkv[jcol];
    const int h    = jcol / 48;        // j = h*48 + dk*3 + c
    const int rem  = jcol % 48;
    const int dk   = rem / 3;
    const int ccls = rem % 3;
    float* dst = (ccls == 0) ? qW : (ccls == 1 ? kW : vW);
    const size_t base = ((size_t)(bn * HH + h) * T) * DK + dk;
    #pragma unroll
    for (int p = 0; p < 8; ++p) {
        int t = tbase + p + 8 * (lane >> 4);
        dst[base + (size_t)t * DK] = c[p] + bias;
    }
}

// ---------------------------------------------------------------------------
// Kernel 2: fused attention + banded edge bias + tv_out + out projection.
// One block per (bn, 16-row t-tile). 256 threads = 8 waves; wave h = head h.
// LDS: 64 KB score buffer Sc[t][s][h]; reused for the xo[t][d] tile.
// ---------------------------------------------------------------------------
__global__ __launch_bounds__(256) void attn_kernel(
    const float* __restrict__ tv, const int* __restrict__ mask,
    const float* __restrict__ local_mask,
    const float* __restrict__ Wg,   const float* __restrict__ bg,
    const float* __restrict__ Wout, const float* __restrict__ bout,
    const float* __restrict__ Wupd, const float* __restrict__ bupd,
    const float* __restrict__ qW, const float* __restrict__ kW,
    const float* __restrict__ vW,
    float* __restrict__ xo_out, float* __restrict__ tv_out)
{
    __shared__ float smem[16 * 128 * 8];    // 64 KB: Sc[t][s][h], later xo[t][d]
    const int tid  = threadIdx.x;
    const int lane = tid & 31;
    const int h    = tid >> 5;              // wave index == head
    const int bn   = blockIdx.x >> 3;
    const int t0   = (blockIdx.x & 7) * 16;
    const int b = bn >> 6, n = bn & 63;
    const int row16 = lane & 15;
    const int khalf = 2 * (lane >> 4);
    const size_t headbase = (size_t)(bn * HH + h) * T * DK;

    // ---- scores = (q k^T) / sqrt(DK), per 16x16 s-tile -------------------
    for (int j = 0; j < 8; ++j) {
        v8f c = {};
        #pragma unroll
        for (int kk = 0; kk < DK; kk += 4) {
            v2f a, bm;
            const float* qrow = qW + headbase + (size_t)(t0 + row16) * DK + kk + khalf;
            a.x = qrow[0]; a.y = qrow[1];
            const float* krow = kW + headbase + (size_t)(j * 16 + row16) * DK + kk + khalf;
            bm.x = krow[0]; bm.y = krow[1];           // B[k][s] = k[s][k]
            c = __builtin_amdgcn_wmma_f32_16x16x4_f32(false, a, false, bm,
                                                      (short)0, c, false, false);
        }
        #pragma unroll
        for (int p = 0; p < 8; ++p) {
            int tl = p + 8 * (lane >> 4);
            int s  = j * 16 + row16;
            smem[(tl * 128 + s) * 8 + h] = c[p] * 0.25f;   // 1/sqrt(16)
        }
    }
    __syncthreads();

    // ---- banded edge bias + clip + key mask (two threads per (h,t) row) --
    {
        int row = tid >> 1;                 // 0..127 -> (h2, tl)
        int h2  = row >> 4;
        int tl  = row & 15;
        int tg  = t0 + tl;
        int s0  = (tid & 1) * 64;
        float bgh = bg[h2];
        for (int s = s0; s < s0 + 64; ++s) {
            int idx = (tl * 128 + s) * 8 + h2;
            float val = smem[idx];
            float lm  = local_mask[tg * T + s];
            if (lm != 0.0f) {               // only |t-s| <= LW reads tv
                const float* tvp = tv + (((size_t)(bn * T + tg)) * T + s) * DE;
                __builtin_prefetch(tvp + DE, 0, 1);   // next band row -> global_prefetch_b8
                float dot = 0.0f;
                #pragma unroll
                for (int de = 0; de < DE; ++de)
                    dot = fmaf(tvp[de], Wg[de * HH + h2], dot);
                val += lm * (dot + bgh);
            }
            val = fminf(fmaxf(val, -5.0f), 5.0f);
            if (mask[b * T + s] == 0) val = 1e-28f;
            smem[idx] = val;
        }
    }
    __syncthreads();

    // ---- tv_out = clipped_scores @ W_upd + b_upd (fused, streamed) -------
    for (int i = 0; i < 8; ++i) {
        int rr = tid + 256 * i;             // (t,s) row id 0..2047
        int tl = rr >> 7;
        int s  = rr & 127;
        float sc[8];
        #pragma unroll
        for (int hh = 0; hh < 8; ++hh) sc[hh] = smem[rr * 8 + hh];
        float* outp = tv_out + (((size_t)(bn * T + t0 + tl)) * T + s) * DE;
        #pragma unroll
        for (int de = 0; de < DE; ++de) {
            float acc = bupd[de];           // weight reads are wave-uniform
            #pragma unroll
            for (int hh = 0; hh < 8; ++hh)
                acc = fmaf(sc[hh], Wupd[hh * DE + de], acc);
            outp[de] = acc;
        }
    }
    __syncthreads();

    // ---- softmax over s (one thread per (h,t) row), then re-mask ---------
    if (tid < 128) {
        int h2 = tid >> 4;
        int tl = tid & 15;
        float mx = -1e30f;
        for (int s = 0; s < 128; ++s)
            mx = fmaxf(mx, smem[(tl * 128 + s) * 8 + h2]);
        float sum = 0.0f;
        for (int s = 0; s < 128; ++s) {
            float e = __expf(smem[(tl * 128 + s) * 8 + h2] - mx);
            smem[(tl * 128 + s) * 8 + h2] = e;
            sum += e;
        }
        float inv = 1.0f / sum;
        for (int s = 0; s < 128; ++s) {
            float a = smem[(tl * 128 + s) * 8 + h2] * inv;
            if (mask[b * T + s] == 0) a = 0.0f;
            smem[(tl * 128 + s) * 8 + h2] = a;
        }
    }
    __syncthreads();

    // ---- xo_h = attn @ v  (16x16 per head, K = 128) ----------------------
    v8f cxo = {};
    #pragma unroll 8
    for (int kk = 0; kk < 128; kk += 4) {
        v2f a, bm;
        a.x = smem[(row16 * 128 + kk + khalf) * 8 + h];
        a.y = smem[(row16 * 128 + kk + khalf + 1) * 8 + h];
        const float* vrow = vW + headbase + (size_t)(kk + khalf) * DK + row16;
        bm.x = vrow[0];                     // B[s][dk] = v[s][dk]
        bm.y = vrow[DK];
        cxo = __builtin_amdgcn_wmma_f32_16x16x4_f32(false, a, false, bm,
                                                    (short)0, cxo, false, false);
    }
    __syncthreads();                        // all attn reads done -> reuse LDS
    #pragma unroll
    for (int p = 0; p < 8; ++p) {
        int tl = p + 8 * (lane >> 4);
        smem[tl * 128 + h * 16 + row16] = cxo[p];    // xo[t][d] tile
    }
    __syncthreads();

    // ---- out = xo @ W_out + b_out  (wave h owns output col tile h) -------
    v8f co = {};
    #pragma unroll 8
    for (int kk = 0; kk < 128; kk += 4) {
        v2f a, bm;
        a.x  = smem[row16 * 128 + kk + khalf];
        a.y  = smem[row16 * 128 + kk + khalf + 1];
        bm.x = Wout[(kk + khalf) * DD + h * 16 + row16];
        bm.y = Wout[(kk + khalf + 1) * DD + h * 16 + row16];
        co = __builtin_amdgcn_wmma_f32_16x16x4_f32(false, a, false, bm,
                                                   (short)0, co, false, false);
    }
    const int dcol = h * 16 + row16;
    const float bo = bout[dcol];
    #pragma unroll
    for (int p = 0; p < 8; ++p) {
        int tg = t0 + p + 8 * (lane >> 4);
        xo_out[(((size_t)(b * T + tg)) * NN + n) * DD + dcol] = co[p] + bo;
    }
}

// ---------------------------------------------------------------------------
extern "C" void kernel_launch(void* const* d_in, const int* in_sizes, int n_in,
                              void* d_out, int out_size, void* d_ws, size_t ws_size,
                              hipStream_t stream) {
    (void)in_sizes; (void)n_in; (void)out_size; (void)ws_size;
    const float* node       = (const float*)d_in[0];
    const float* tv         = (const float*)d_in[1];
    const int*   mask       = (const int*)  d_in[2];
    const float* local_mask = (const float*)d_in[3];
    const float* Wqkv       = (const float*)d_in[4];
    const float* bqkv       = (const float*)d_in[5];
    const float* Wg         = (const float*)d_in[6];
    const float* bg         = (const float*)d_in[7];
    const float* Wout       = (const float*)d_in[8];
    const float* bout       = (const float*)d_in[9];
    const float* Wupd       = (const float*)d_in[10];
    const float* bupd       = (const float*)d_in[11];

    // workspace: q/k/v each (BN,H,T,DK) fp32 = 8 MB  -> 24 MB total
    const size_t qkv_elems = (size_t)(BB * NN) * HH * T * DK;
    float* qW = (float*)d_ws;
    float* kW = qW + qkv_elems;
    float* vW = kW + qkv_elems;

    float* xo_out = (float*)d_out;                            // (B,T,N,D)
    float* tvout  = xo_out + (size_t)BB * T * NN * DD;        // (B,N,T,T,DE)

    qkv_kernel<<<dim3(6, 1024), 128, 0, stream>>>(node, Wqkv, bqkv, qW, kW, vW);
    attn_kernel<<<dim3(1024), 256, 0, stream>>>(tv, mask, local_mask, Wg, bg,
                                                Wout, bout, Wupd, bupd,
                                                qW, kW, vW, xo_out, tvout);
}